// RoIAlign_27539330302287
// MI455X (gfx1250) — compile-verified
//
#include <hip/hip_runtime.h>
#include <stdint.h>

// RoIAlign, NCHW fp32, spatial_scale=0.25, out 7x7, sample_num=2.
// features: (2, 256, 200, 336) f32; rois: (N=1024, 5) f32 [b, x1, y1, x2, y2]
// out: (N, 256, 7, 7) f32
//
// Memory-bound gather: HBM floor ~190MB (~8us @ 23.3TB/s); features (137.6MB)
// are L2-resident (192MB), so the ~820MB of corner-gather traffic is served
// from L2. WMMA does not apply (per-sample data-dependent indices AND weights
// -> no shared-operand matmul). CDNA5 paths used:
//   - wave32 blocks, SALU-uniform ROI math, LDS broadcast descriptors
//   - global_prefetch_b8 row warming
//   - TENSOR_STORE_FROM_LDS (TDM DMA) for the contiguous 49KB tile writeback,
//     issued by one wave, synced with s_wait_tensorcnt.

#define OUT_HW   7
#define NS       2
#define C_CH     256
#define FH       200
#define FW       336
#define SSCALE   0.25f
#define NPIX     (OUT_HW * OUT_HW)          // 49
#define NSAMP    (OUT_HW * NS)              // 14 per axis
#define NELEM    (C_CH * NPIX)              // 12544 floats per ROI tile

typedef unsigned int u32x4 __attribute__((ext_vector_type(4)));
typedef int          i32x4 __attribute__((ext_vector_type(4)));
typedef int          i32x8 __attribute__((ext_vector_type(8)));

struct YDesc { int yoff; int dy; float w0; float w1; };   // row elem offset, clamped drow, (1-ly)*ym*0.25, ly*ym*0.25
struct XDesc { int xoff; float wl; float wr; int pad; };  // clamped col (<=W-2), pair weights with xmask folded

__global__ __launch_bounds__(C_CH) void roi_align_f32_kernel(
    const float* __restrict__ features,
    const float* __restrict__ rois,
    float* __restrict__ out)
{
    __shared__ YDesc sy[NSAMP];
    __shared__ XDesc sx[NSAMP];
    __shared__ float sout[NELEM];           // 49KB staging tile

    const int n = blockIdx.x;
    const int t = threadIdx.x;
    const float* __restrict__ roi = rois + (size_t)n * 5;

    // ---- Phase 1: separable sample descriptors (28 threads) ----
    if (t < 2 * NSAMP) {
        const float x1 = roi[1] * SSCALE;
        const float y1 = roi[2] * SSCALE;
        const float x2 = roi[3] * SSCALE;
        const float y2 = roi[4] * SSCALE;
        if (t < NSAMP) {
            // y-axis: fold ymask and the 1/4 sample-average into weights
            const float roi_h = fmaxf(y2 - y1, 1.0f);
            const float step  = roi_h * (1.0f / (OUT_HW * NS));     // bin_h / ns
            const float yc    = y1 + ((float)t + 0.5f) * step;
            const float m     = (yc >= -1.0f && yc <= (float)FH) ? 0.25f : 0.0f;
            const float y     = fminf(fmaxf(yc, 0.0f), (float)(FH - 1));
            const int   y0    = (int)floorf(y);
            const float ly    = y - (float)y0;
            const int   y1i   = min(y0 + 1, FH - 1);
            sy[t].yoff = y0 * FW;
            sy[t].dy   = (y1i - y0) * FW;
            sy[t].w0   = (1.0f - ly) * m;
            sy[t].w1   = ly * m;
        } else {
            // x-axis: clamp column so [xoff, xoff+1] is a valid in-row pair;
            // shift weights when x0 == W-1 (there lx == 0).
            const int   q     = t - NSAMP;
            const float roi_w = fmaxf(x2 - x1, 1.0f);
            const float step  = roi_w * (1.0f / (OUT_HW * NS));
            const float xc    = x1 + ((float)q + 0.5f) * step;
            const float m     = (xc >= -1.0f && xc <= (float)FW) ? 1.0f : 0.0f;
            const float x     = fminf(fmaxf(xc, 0.0f), (float)(FW - 1));
            const int   x0    = (int)floorf(x);
            const float lx    = x - (float)x0;
            const bool  shift = (x0 >= FW - 1);
            sx[q].xoff = shift ? (FW - 2) : x0;
            sx[q].wl   = (shift ? 0.0f : (1.0f - lx)) * m;
            sx[q].wr   = (shift ? 1.0f : lx) * m;
            sx[q].pad  = 0;
        }
    }
    __syncthreads();

    // ---- Phase 2: per-channel gather + bilinear combine ----
    const int c = t;
    const int b = (int)roi[0];
    const float* __restrict__ fp =
        features + (size_t)(b * C_CH + c) * (size_t)(FH * FW);

    for (int ph = 0; ph < OUT_HW; ++ph) {
        const YDesc ya = sy[2 * ph];
        const YDesc yb = sy[2 * ph + 1];
        const float* __restrict__ r0a = fp + ya.yoff;
        const float* __restrict__ r1a = r0a + ya.dy;
        const float* __restrict__ r0b = fp + yb.yoff;
        const float* __restrict__ r1b = r0b + yb.dy;

        // Warm L2/L0 for the next bin-row (global_prefetch_b8 on gfx1250).
        if (ph + 1 < OUT_HW) {
            __builtin_prefetch(fp + sy[2 * ph + 2].yoff, 0, 2);
            __builtin_prefetch(fp + sy[2 * ph + 3].yoff, 0, 2);
        }

        for (int pw = 0; pw < OUT_HW; ++pw) {
            const XDesc xa = sx[2 * pw];
            const XDesc xb = sx[2 * pw + 1];

            float acc;
            acc  = ya.w0 * (xa.wl * r0a[xa.xoff] + xa.wr * r0a[xa.xoff + 1])
                 + ya.w1 * (xa.wl * r1a[xa.xoff] + xa.wr * r1a[xa.xoff + 1]);
            acc += ya.w0 * (xb.wl * r0a[xb.xoff] + xb.wr * r0a[xb.xoff + 1])
                 + ya.w1 * (xb.wl * r1a[xb.xoff] + xb.wr * r1a[xb.xoff + 1]);
            acc += yb.w0 * (xa.wl * r0b[xa.xoff] + xa.wr * r0b[xa.xoff + 1])
                 + yb.w1 * (xa.wl * r1b[xa.xoff] + xa.wr * r1b[xa.xoff + 1]);
            acc += yb.w0 * (xb.wl * r0b[xb.xoff] + xb.wr * r0b[xb.xoff + 1])
                 + yb.w1 * (xb.wl * r1b[xb.xoff] + xb.wr * r1b[xb.xoff + 1]);

            // stride 49 (odd) across lanes -> conflict-free LDS banks
            sout[c * NPIX + ph * OUT_HW + pw] = acc;
        }
    }
    __syncthreads();

    // ---- Phase 3: tile writeback, (C,7,7) contiguous = 50176 bytes ----
#if defined(__gfx1250__) && __has_builtin(__builtin_amdgcn_tensor_store_from_lds)
    // Single TDM DMA per block: LDS -> global, 1-D tile of 12544 x 4B.
    // D# per CDNA5 ISA (groups 0/1; 2/3 unused for <=2D).
    if (t < 32) {   // wave 0 issues; TDM ignores EXEC, descriptor is uniform
        const unsigned long long ga =
            (unsigned long long)(uintptr_t)(out + (size_t)n * NELEM);
        const unsigned int lds_off = (unsigned int)(uintptr_t)(&sout[0]);

        u32x4 g0;
        g0.x = 1u;                                        // count=1, user desc
        g0.y = lds_off;                                   // lds_addr (bytes)
        g0.z = (unsigned int)(ga & 0xffffffffu);          // global_addr[31:0]
        g0.w = (unsigned int)((ga >> 32) & 0x01ffffffu)   // global_addr[56:32]
             | (2u << 30);                                // type = 2 ("image")

        i32x8 g1;
        g1[0] = (int)(2u << 16);                          // data_size=2 -> 4B
        g1[1] = (int)((NELEM & 0xffffu) << 16);           // tensor_dim0[15:0]
        g1[2] = (int)(((NELEM >> 16) & 0xffffu)           // tensor_dim0[31:16]
             |  (1u << 16));                              // tensor_dim1 = 1
        g1[3] = (int)((NELEM & 0xffffu) << 16);           // tile_dim0 = 12544
        g1[4] = 0;                                        // tile_dim1/2 unused
        g1[5] = (int)NELEM;                               // dim0_stride[31:0]
        g1[6] = 0;                                        // stride msbs
        g1[7] = 0;

        i32x4 g2 = {0, 0, 0, 0};
        i32x4 g3 = {0, 0, 0, 0};
#if __clang_major__ >= 23
        i32x8 g4 = {0, 0, 0, 0, 0, 0, 0, 0};
        __builtin_amdgcn_tensor_store_from_lds(g0, g1, g2, g3, g4, 0);
#else
        __builtin_amdgcn_tensor_store_from_lds(g0, g1, g2, g3, 0);
#endif
    }
    // Hold the workgroup's LDS alive until the DMA drains.
    __builtin_amdgcn_s_wait_tensorcnt(0);
#else
    float* __restrict__ outp = out + (size_t)n * NELEM;
    #pragma unroll
    for (int i = 0; i < NPIX; ++i) {                 // 12544 / 256 = 49 iters
        const int idx = t + i * C_CH;
        outp[idx] = sout[idx];
    }
#endif
}

extern "C" void kernel_launch(void* const* d_in, const int* in_sizes, int n_in,
                              void* d_out, int out_size, void* d_ws, size_t ws_size,
                              hipStream_t stream) {
    (void)n_in; (void)d_ws; (void)ws_size; (void)out_size;
    const float* features = (const float*)d_in[0];
    const float* rois     = (const float*)d_in[1];
    float*       out      = (float*)d_out;

    const int N = in_sizes[1] / 5;   // 1024 ROIs
    roi_align_f32_kernel<<<N, C_CH, 0, stream>>>(features, rois, out);
}